// cluster_loss_helper_88785563943727
// MI455X (gfx1250) — compile-verified
//
#include <hip/hip_runtime.h>
#include <hip/hip_bf16.h>

typedef __attribute__((ext_vector_type(2))) float v2f;
typedef __attribute__((ext_vector_type(8))) float v8f;

#define NB   1024        // streaming blocks
#define NT   256         // threads per block (8 wave32)
#define KL   5           // labels
#define NACC 25          // 20 channel-sums (k*4+c) + 5 counts

// workspace layout (float offsets)
#define WS_PARTA   0                    // [25][NB]
#define WS_PARTC   (NACC * NB)          // [5][NB]
#define WS_COUNTS  (WS_PARTC + KL * NB) // [5]
#define WS_MUC     (WS_COUNTS + 8)      // [5][4]
#define WS_LDIST   (WS_MUC + 24)        // scalar

__device__ __forceinline__ float wave_sum32(float v) {
    v += __shfl_xor(v, 16, 32);
    v += __shfl_xor(v, 8, 32);
    v += __shfl_xor(v, 4, 32);
    v += __shfl_xor(v, 2, 32);
    v += __shfl_xor(v, 1, 32);
    return v;
}

// ---------------- Pass A: per-block partial centroid sums/counts -------------
__global__ __launch_bounds__(NT) void k_centroid_partials(
    const float* __restrict__ pred, const int* __restrict__ lab,
    float* __restrict__ partA, int HW)
{
    float acc[NACC];
#pragma unroll
    for (int i = 0; i < NACC; ++i) acc[i] = 0.f;

    const int tid     = blockIdx.x * NT + threadIdx.x;
    const int stride  = NB * NT;
    const int ngroups = HW >> 2;                 // 4 pixels per iteration
    const float* p0 = pred;
    const float* p1 = pred + HW;
    const float* p2 = pred + 2 * HW;
    const float* p3 = pred + 3 * HW;

    for (int g = tid; g < ngroups; g += stride) {
        const int p = g << 2;
        if (g + stride < ngroups) {              // gfx1250 global_prefetch_b8
            const int pn = p + (stride << 2);
            __builtin_prefetch(p0 + pn, 0, 0);
            __builtin_prefetch(p1 + pn, 0, 0);
            __builtin_prefetch(p2 + pn, 0, 0);
            __builtin_prefetch(p3 + pn, 0, 0);
            __builtin_prefetch(lab + pn, 0, 0);
        }
        const int4   l4 = *(const int4*)(lab + p);
        const float4 a  = *(const float4*)(p0 + p);
        const float4 b  = *(const float4*)(p1 + p);
        const float4 c  = *(const float4*)(p2 + p);
        const float4 d  = *(const float4*)(p3 + p);
        const int   ls[4]    = {l4.x, l4.y, l4.z, l4.w};
        const float xs[4][4] = {{a.x, b.x, c.x, d.x}, {a.y, b.y, c.y, d.y},
                                {a.z, b.z, c.z, d.z}, {a.w, b.w, c.w, d.w}};
#pragma unroll
        for (int j = 0; j < 4; ++j) {
#pragma unroll
            for (int k = 0; k < KL; ++k) {
                const float sel = (ls[j] == k) ? 1.f : 0.f;
                acc[k * 4 + 0] = fmaf(sel, xs[j][0], acc[k * 4 + 0]);
                acc[k * 4 + 1] = fmaf(sel, xs[j][1], acc[k * 4 + 1]);
                acc[k * 4 + 2] = fmaf(sel, xs[j][2], acc[k * 4 + 2]);
                acc[k * 4 + 3] = fmaf(sel, xs[j][3], acc[k * 4 + 3]);
                acc[20 + k]   += sel;
            }
        }
    }

    __shared__ float red[NT / 32][NACC];
    const int wave = threadIdx.x >> 5, lane = threadIdx.x & 31;
#pragma unroll
    for (int i = 0; i < NACC; ++i) acc[i] = wave_sum32(acc[i]);
    if (lane == 0) {
#pragma unroll
        for (int i = 0; i < NACC; ++i) red[wave][i] = acc[i];
    }
    __syncthreads();
    if (threadIdx.x < NACC) {
        float s = 0.f;
#pragma unroll
        for (int w = 0; w < NT / 32; ++w) s += red[w][threadIdx.x];
        partA[threadIdx.x * NB + blockIdx.x] = s;
    }
}

// -------- Finalize centroids + L_dist via V_WMMA_F32_16X16X4_F32 Gram -------
__global__ __launch_bounds__(NT) void k_finalize_centroids(
    const float* __restrict__ partA, const float* __restrict__ delta_d_p,
    float* __restrict__ ws)
{
    __shared__ float tot[32];
    __shared__ float mucp[16][4];     // centroids zero-padded to 16 rows
    __shared__ float gram[16][16];
    const int tid = threadIdx.x;

    // reduce [25][NB] partials: 8 threads per value
    const int v = tid >> 3, sub = tid & 7;
    float s = 0.f;
    if (v < NACC) {
        const float* col = partA + v * NB;
        for (int i = sub; i < NB; i += 8) s += col[i];
    }
    s += __shfl_xor(s, 4, 32);
    s += __shfl_xor(s, 2, 32);
    s += __shfl_xor(s, 1, 32);
    if (sub == 0 && v < NACC) tot[v] = s;
    __syncthreads();

    if (tid < 64) {                               // muc = sums / max(count,1)
        const int r = tid >> 2, c = tid & 3;
        float m = 0.f;
        if (r < KL) {
            m = tot[r * 4 + c] / fmaxf(tot[20 + r], 1.f);
            ws[WS_MUC + r * 4 + c] = m;
        }
        mucp[r][c] = m;
    }
    if (tid < KL) ws[WS_COUNTS + tid] = tot[20 + tid];
    __syncthreads();

    if (tid < 32) {                               // full wave, EXEC all ones
        const int lane = tid;
        const int m  = lane & 15;
        const int kh = (lane < 16) ? 0 : 2;       // 16x4 f32 A-matrix layout
        v2f A; A.x = mucp[m][kh]; A.y = mucp[m][kh + 1];
        v2f B; B.x = A.x;         B.y = A.y;      // B = A^T of same values (Gram)
        v8f C = {0.f, 0.f, 0.f, 0.f, 0.f, 0.f, 0.f, 0.f};
        v8f D = __builtin_amdgcn_wmma_f32_16x16x4_f32(
            /*neg_a=*/false, A, /*neg_b=*/false, B,
            /*c_mod=*/(short)0, C, /*reuse_a=*/false, /*reuse_b=*/false);
#pragma unroll
        for (int r = 0; r < 8; ++r)               // 16x16 f32 C/D layout
            gram[r + ((lane < 16) ? 0 : 8)][m] = D[r];
    }
    __syncthreads();

    if (tid == 0) {
        const float dd = delta_d_p[0];
        float L = 0.f;
        for (int a = 0; a < KL; ++a)
            for (int b = 0; b < KL; ++b) {
                if (a == b) continue;
                const float csq = gram[a][a] + gram[b][b] - 2.f * gram[a][b];
                const float cd  = csq > 0.f ? sqrtf(csq) : 0.f;
                const float h   = fmaxf(dd - cd, 0.f);
                L = fmaf(h, h, L);
            }
        ws[WS_LDIST] = L / (float)(KL * (KL - 1));
    }
}

// ---------------- Pass C: per-block partial hinge-variance sums --------------
__global__ __launch_bounds__(NT) void k_var_partials(
    const float* __restrict__ pred, const int* __restrict__ lab,
    const float* __restrict__ ws, const float* __restrict__ delta_v_p,
    float* __restrict__ partC, int HW)
{
    __shared__ float4 smuc[KL];
    if (threadIdx.x < KL) {
        const float* m = ws + WS_MUC + threadIdx.x * 4;
        smuc[threadIdx.x] = make_float4(m[0], m[1], m[2], m[3]);
    }
    __syncthreads();
    const float dv = delta_v_p[0];

    float vs[KL] = {0.f, 0.f, 0.f, 0.f, 0.f};
    const int tid     = blockIdx.x * NT + threadIdx.x;
    const int stride  = NB * NT;
    const int ngroups = HW >> 2;
    const float* p0 = pred;
    const float* p1 = pred + HW;
    const float* p2 = pred + 2 * HW;
    const float* p3 = pred + 3 * HW;

    for (int g = tid; g < ngroups; g += stride) {
        const int p = g << 2;
        if (g + stride < ngroups) {
            const int pn = p + (stride << 2);
            __builtin_prefetch(p0 + pn, 0, 0);
            __builtin_prefetch(p1 + pn, 0, 0);
            __builtin_prefetch(p2 + pn, 0, 0);
            __builtin_prefetch(p3 + pn, 0, 0);
            __builtin_prefetch(lab + pn, 0, 0);
        }
        const int4   l4 = *(const int4*)(lab + p);
        const float4 a  = *(const float4*)(p0 + p);
        const float4 b  = *(const float4*)(p1 + p);
        const float4 c  = *(const float4*)(p2 + p);
        const float4 d  = *(const float4*)(p3 + p);
        const int   ls[4]    = {l4.x, l4.y, l4.z, l4.w};
        const float xs[4][4] = {{a.x, b.x, c.x, d.x}, {a.y, b.y, c.y, d.y},
                                {a.z, b.z, c.z, d.z}, {a.w, b.w, c.w, d.w}};
#pragma unroll
        for (int j = 0; j < 4; ++j) {
            const float4 mu = smuc[ls[j]];        // single ds_load_b128 gather
            const float d0 = mu.x - xs[j][0];
            const float d1 = mu.y - xs[j][1];
            const float d2 = mu.z - xs[j][2];
            const float d3 = mu.w - xs[j][3];
            const float dsq = fmaf(d0, d0, fmaf(d1, d1, fmaf(d2, d2, d3 * d3)));
            const float h   = fmaxf(sqrtf(dsq) - dv, 0.f);
            const float var = h * h;
#pragma unroll
            for (int k = 0; k < KL; ++k)
                vs[k] += (ls[j] == k) ? var : 0.f;
        }
    }

    __shared__ float red[NT / 32][KL];
    const int wave = threadIdx.x >> 5, lane = threadIdx.x & 31;
#pragma unroll
    for (int k = 0; k < KL; ++k) vs[k] = wave_sum32(vs[k]);
    if (lane == 0) {
#pragma unroll
        for (int k = 0; k < KL; ++k) red[wave][k] = vs[k];
    }
    __syncthreads();
    if (threadIdx.x < KL) {
        float s = 0.f;
#pragma unroll
        for (int w = 0; w < NT / 32; ++w) s += red[w][threadIdx.x];
        partC[threadIdx.x * NB + blockIdx.x] = s;
    }
}

// ---------------- Finalize: L_var/K + L_dist --------------------------------
__global__ __launch_bounds__(NT) void k_finalize_loss(
    const float* __restrict__ partC, const float* __restrict__ ws,
    float* __restrict__ out)
{
    __shared__ float lv[8];
    const int v = threadIdx.x >> 5, lane = threadIdx.x & 31;   // wave per label
    float s = 0.f;
    if (v < KL) {
        const float* col = partC + v * NB;
        for (int i = lane; i < NB; i += 32) s += col[i];
    }
    s = wave_sum32(s);
    if (lane == 0) lv[v] = s;
    __syncthreads();
    if (threadIdx.x == 0) {
        float L_var = 0.f;
        for (int k = 0; k < KL; ++k)
            L_var += lv[k] / fmaxf(ws[WS_COUNTS + k], 1.f);
        L_var /= (float)KL;
        out[0] = L_var + ws[WS_LDIST];
    }
}

extern "C" void kernel_launch(void* const* d_in, const int* in_sizes, int n_in,
                              void* d_out, int out_size, void* d_ws, size_t ws_size,
                              hipStream_t stream)
{
    const float* pred = (const float*)d_in[0];
    const int*   lab  = (const int*)d_in[1];
    const float* dv   = (const float*)d_in[2];
    const float* dd   = (const float*)d_in[3];
    float* ws  = (float*)d_ws;
    float* out = (float*)d_out;
    const int HW = in_sizes[1];                  // 1024*2048

    k_centroid_partials<<<NB, NT, 0, stream>>>(pred, lab, ws + WS_PARTA, HW);
    k_finalize_centroids<<<1, NT, 0, stream>>>(ws + WS_PARTA, dd, ws);
    k_var_partials<<<NB, NT, 0, stream>>>(pred, lab, ws, dv, ws + WS_PARTC, HW);
    k_finalize_loss<<<1, NT, 0, stream>>>(ws + WS_PARTC, ws, out);
}